// RelationalLayer_17746804867119
// MI455X (gfx1250) — compile-verified
//
#include <hip/hip_runtime.h>

typedef float v2f __attribute__((ext_vector_type(2)));
typedef float v8f __attribute__((ext_vector_type(8)));

#define N_AGENT 81
#define N_NBR   20
#define DD      96
#define HH      96
#define ROWS_PAD 96   // 81 agents padded to 6 full 16-row WMMA tiles

__global__ __launch_bounds__(128) void relational_wmma_kernel(
    const float* __restrict__ x,      // [512,81,96]
    const int*   __restrict__ edges,  // [81,20]
    const float* __restrict__ W1,     // [192,96]
    const float* __restrict__ b1,     // [96]
    const float* __restrict__ W2,     // [96,96]
    const float* __restrict__ b2,     // [96]
    float* __restrict__ out)          // [512,81,96]
{
    extern __shared__ float lds[];
    float* xh   = lds;                     // [96][96] x tile; later reused as hsum
    float* sbuf = lds + 96 * 96;           // [96][96] s = x @ W1_top
    float* ybuf = lds + 2 * 96 * 96;       // [96][96] y = x @ W1_bot
    int*   eg   = (int*)(lds + 3 * 96 * 96); // [81*20]

    const int b    = blockIdx.x;
    const int tid  = threadIdx.x;
    const int lane = tid & 31;
    const int wave = tid >> 5;
    const int lh   = (lane >> 4) & 1;      // lane half selects K pair
    const int l15  = lane & 15;            // M (A-frag) / N (B,C,D frags)

    // ---- Phase 1: stage x[b] (zero-pad rows 81..95) and edges into LDS ----
    const float* xb = x + (size_t)b * (N_AGENT * DD);
    for (int f = tid; f < (ROWS_PAD * DD) / 4; f += 128) {
        int row = f / (DD / 4);
        int c4  = f % (DD / 4);
        float4 v = make_float4(0.f, 0.f, 0.f, 0.f);
        if (row < N_AGENT) v = ((const float4*)(xb + row * DD))[c4];
        ((float4*)(xh + row * DD))[c4] = v;
    }
    for (int f = tid; f < N_AGENT * N_NBR; f += 128) eg[f] = edges[f];
    __syncthreads();

    // ---- Phase 2: s = x @ W1[0:96,:], y = x @ W1[96:192,:]  (WMMA f32 16x16x4)
    // 6 m-tiles x 6 n-tiles = 36 tiles over 4 waves; A-fragment shared by both GEMMs.
    for (int t = wave; t < 36; t += 4) {
        const int mt = t / 6, nt = t % 6;
        v8f accS = {};
        v8f accY = {};
        const int m = mt * 16 + l15;
        const int n = nt * 16 + l15;
        for (int k = 0; k < DD; k += 4) {
            const int k0 = k + lh * 2;
            v2f a;
            a.x = xh[m * DD + k0];
            a.y = xh[m * DD + k0 + 1];
            v2f bt, bb;
            bt.x = W1[(k0)*HH + n];
            bt.y = W1[(k0 + 1) * HH + n];
            bb.x = W1[(DD + k0) * HH + n];
            bb.y = W1[(DD + k0 + 1) * HH + n];
            accS = __builtin_amdgcn_wmma_f32_16x16x4_f32(false, a, false, bt,
                                                         (short)0, accS, false, false);
            accY = __builtin_amdgcn_wmma_f32_16x16x4_f32(false, a, false, bb,
                                                         (short)0, accY, false, false);
        }
        for (int j = 0; j < 8; ++j) {
            const int mr = mt * 16 + j + lh * 8;   // C/D layout: VGPR j -> M=j (+8 for hi half)
            sbuf[mr * HH + n] = accS[j];
            ybuf[mr * HH + n] = accY[j];
        }
    }
    __syncthreads();

    // ---- Phase 3: hsum[a][c] = sum_n relu(s[a][c] + b1[c] + y[edges[a,n]][c])
    // hsum overwrites xh rows 0..80 (rows 81..95 remain the zero padding).
    for (int idx = tid; idx < N_AGENT * HH; idx += 128) {
        const int a = idx / HH, c = idx % HH;
        const float sa = sbuf[a * HH + c] + b1[c];
        float v = 0.f;
        #pragma unroll 5
        for (int nn = 0; nn < N_NBR; ++nn) {
            const int j = eg[a * N_NBR + nn];
            const float hv = sa + ybuf[j * HH + c];
            v += fmaxf(hv, 0.f);
        }
        xh[a * HH + c] = v;
    }
    __syncthreads();

    // ---- Phase 4: out[b] = hsum @ W2 + 20*b2 ----
    float* ob = out + (size_t)b * (N_AGENT * HH);
    for (int t = wave; t < 36; t += 4) {
        const int mt = t / 6, nt = t % 6;
        v8f acc = {};
        const int m = mt * 16 + l15;
        const int n = nt * 16 + l15;
        for (int k = 0; k < HH; k += 4) {
            const int k0 = k + lh * 2;
            v2f a, bw;
            a.x  = xh[m * HH + k0];
            a.y  = xh[m * HH + k0 + 1];
            bw.x = W2[(k0)*HH + n];
            bw.y = W2[(k0 + 1) * HH + n];
            acc = __builtin_amdgcn_wmma_f32_16x16x4_f32(false, a, false, bw,
                                                        (short)0, acc, false, false);
        }
        const float bias = (float)N_NBR * b2[n];
        for (int j = 0; j < 8; ++j) {
            const int mr = mt * 16 + j + lh * 8;
            if (mr < N_AGENT) ob[mr * HH + n] = acc[j] + bias;
        }
    }
}

extern "C" void kernel_launch(void* const* d_in, const int* in_sizes, int n_in,
                              void* d_out, int out_size, void* d_ws, size_t ws_size,
                              hipStream_t stream) {
    (void)in_sizes; (void)n_in; (void)out_size; (void)d_ws; (void)ws_size;
    const float* x     = (const float*)d_in[0];
    const int*   edges = (const int*)d_in[1];
    const float* W1    = (const float*)d_in[2];
    const float* b1    = (const float*)d_in[3];
    const float* W2    = (const float*)d_in[4];
    const float* b2    = (const float*)d_in[5];
    float* out = (float*)d_out;

    const size_t shmem = (size_t)(3 * 96 * 96) * sizeof(float)
                       + (size_t)(N_AGENT * N_NBR) * sizeof(int); // 117072 B
    relational_wmma_kernel<<<dim3(512), dim3(128), shmem, stream>>>(
        x, edges, W1, b1, W2, b2, out);
}